// NGCFLayer_30751965840097
// MI455X (gfx1250) — compile-verified
//
#include <hip/hip_runtime.h>
#include <hip/hip_bf16.h>

typedef float f32x2 __attribute__((ext_vector_type(2)));
typedef float f32x8 __attribute__((ext_vector_type(8)));

#define D 128

// =====================================================================
// Kernel 1: P1 = ego @ W1   (N x 128) = (N x 128)(128 x 128), fp32 WMMA
// One wave per 16-row tile, 8 waves per block. W1 staged in LDS (64 KB).
// =====================================================================
__global__ void __launch_bounds__(256)
ngcf_p1_gemm(const float* __restrict__ ego, const float* __restrict__ W1,
             float* __restrict__ P1, int N)
{
    __shared__ float sW[D * D];
    for (int i = threadIdx.x; i < D * D / 4; i += 256)
        ((float4*)sW)[i] = ((const float4*)W1)[i];
    __syncthreads();

    const int wave = threadIdx.x >> 5;
    const int lane = threadIdx.x & 31;
    const int half = lane >> 4;   // 0: lanes 0-15, 1: lanes 16-31
    const int l16  = lane & 15;

    const int tile    = blockIdx.x * 8 + wave;
    const int rowBase = tile * 16;
    if (rowBase >= N) return;

    f32x8 acc[8];
#pragma unroll
    for (int n = 0; n < 8; ++n)
#pragma unroll
        for (int j = 0; j < 8; ++j) acc[n][j] = 0.0f;

    // A fragment source row: M = lane % 16 (both halves), K offset 2*half.
    int arow_idx = rowBase + l16;
    if (arow_idx >= N) arow_idx = N - 1;            // clamp; stores guarded
    const float* __restrict__ arow = ego + (size_t)arow_idx * D;

    for (int k0 = 0; k0 < D; k0 += 4) {
        const int k = k0 + 2 * half;
        f32x2 a;
        a.x = arow[k];
        a.y = arow[k + 1];
#pragma unroll
        for (int n = 0; n < 8; ++n) {
            const int col = n * 16 + l16;
            f32x2 b;
            b.x = sW[k * D + col];
            b.y = sW[(k + 1) * D + col];
            acc[n] = __builtin_amdgcn_wmma_f32_16x16x4_f32(
                false, a, false, b, (short)0, acc[n], false, false);
        }
    }

    // D layout: VGPR v, lane l -> row v + 8*half, col l16
#pragma unroll
    for (int v = 0; v < 8; ++v) {
        const int row = rowBase + v + 8 * half;
        if (row < N) {
#pragma unroll
            for (int n = 0; n < 8; ++n)
                P1[(size_t)row * D + n * 16 + l16] = acc[n][v];
        }
    }
}

// =====================================================================
// Kernel 2: per-edge messages.
//   t    = h_src * h_dst            (elementwise, gathered, L2-resident)
//   msg  = P1[src] + t @ W2         (WMMA fp32, W2 staged in LDS)
//   out[dst] += msg * norm[src]*norm[dst]   (global_atomic_add_f32)
// One wave per 16-edge tile, 8 waves per block.
// =====================================================================
__global__ void __launch_bounds__(256)
ngcf_edge_kernel(const float* __restrict__ ego, const float* __restrict__ norm,
                 const int* __restrict__ src, const int* __restrict__ dst,
                 const float* __restrict__ W2, const float* __restrict__ P1,
                 float* __restrict__ out, int E)
{
    __shared__ float sW[D * D];
    for (int i = threadIdx.x; i < D * D / 4; i += 256)
        ((float4*)sW)[i] = ((const float4*)W2)[i];
    __syncthreads();

    const int wave = threadIdx.x >> 5;
    const int lane = threadIdx.x & 31;
    const int half = lane >> 4;
    const int l16  = lane & 15;

    const int tile     = blockIdx.x * 8 + wave;
    const int edgeBase = tile * 16;
    if (edgeBase >= E) return;

    // A-matrix row for this lane: edge = edgeBase + l16 (clamped to 0 past E;
    // its contribution is killed by scale=0 at scatter time).
    const int  eA   = edgeBase + l16;
    const bool vA   = (eA < E);
    const int  sIdx = vA ? src[eA] : 0;
    const int  dIdx = vA ? dst[eA] : 0;
    const float* __restrict__ hs = ego + (size_t)sIdx * D;
    const float* __restrict__ hd = ego + (size_t)dIdx * D;

    f32x8 acc[8];
#pragma unroll
    for (int n = 0; n < 8; ++n)
#pragma unroll
        for (int j = 0; j < 8; ++j) acc[n][j] = 0.0f;

    for (int k0 = 0; k0 < D; k0 += 4) {
        const int k = k0 + 2 * half;
        f32x2 a;
        a.x = hs[k]     * hd[k];
        a.y = hs[k + 1] * hd[k + 1];
#pragma unroll
        for (int n = 0; n < 8; ++n) {
            const int col = n * 16 + l16;
            f32x2 b;
            b.x = sW[k * D + col];
            b.y = sW[(k + 1) * D + col];
            acc[n] = __builtin_amdgcn_wmma_f32_16x16x4_f32(
                false, a, false, b, (short)0, acc[n], false, false);
        }
    }

    // Scatter: VGPR v, lane l holds msg[edgeBase + v + 8*half][n*16 + l16]
#pragma unroll
    for (int v = 0; v < 8; ++v) {
        const int e = edgeBase + v + 8 * half;
        int   s = 0, d = 0;
        float scale = 0.0f;
        if (e < E) {
            s = src[e];
            d = dst[e];
            scale = norm[s] * norm[d];
        }
        const float* __restrict__ p1row = P1 + (size_t)s * D;
        float* __restrict__ orow = out + (size_t)d * D;
#pragma unroll
        for (int n = 0; n < 8; ++n) {
            const int col = n * 16 + l16;
            const float m = (acc[n][v] + p1row[col]) * scale;
            atomicAdd(&orow[col], m);
        }
    }
}

// =====================================================================
// Kernel 3: finalize.  h = leakyrelu(out + P1); out = h / max(||h||, 1e-12)
// One wave per row (32 lanes x float4 = 128 cols), 8 rows per block.
// =====================================================================
__global__ void __launch_bounds__(256)
ngcf_finalize(float* __restrict__ out, const float* __restrict__ P1, int N)
{
    const int row  = blockIdx.x * 8 + (threadIdx.x >> 5);
    const int lane = threadIdx.x & 31;
    if (row >= N) return;

    float4 x = ((const float4*)(out + (size_t)row * D))[lane];
    float4 p = ((const float4*)(P1  + (size_t)row * D))[lane];
    float v0 = x.x + p.x, v1 = x.y + p.y, v2 = x.z + p.z, v3 = x.w + p.w;
    v0 = (v0 >= 0.0f) ? v0 : 0.2f * v0;
    v1 = (v1 >= 0.0f) ? v1 : 0.2f * v1;
    v2 = (v2 >= 0.0f) ? v2 : 0.2f * v2;
    v3 = (v3 >= 0.0f) ? v3 : 0.2f * v3;

    float ss = v0 * v0 + v1 * v1 + v2 * v2 + v3 * v3;
#pragma unroll
    for (int off = 16; off > 0; off >>= 1)
        ss += __shfl_xor(ss, off, 32);

    const float inv = 1.0f / fmaxf(sqrtf(ss), 1e-12f);
    float4 r;
    r.x = v0 * inv; r.y = v1 * inv; r.z = v2 * inv; r.w = v3 * inv;
    ((float4*)(out + (size_t)row * D))[lane] = r;
}

// =====================================================================
// Launcher
// =====================================================================
extern "C" void kernel_launch(void* const* d_in, const int* in_sizes, int n_in,
                              void* d_out, int out_size, void* d_ws, size_t ws_size,
                              hipStream_t stream) {
    const float* ego  = (const float*)d_in[0];   // [N,128]
    const float* norm = (const float*)d_in[1];   // [N,1]
    const int*   src  = (const int*)d_in[2];     // [E]
    const int*   dst  = (const int*)d_in[3];     // [E]
    const float* W1   = (const float*)d_in[4];   // [128,128]
    const float* W2   = (const float*)d_in[5];   // [128,128]
    float* out = (float*)d_out;                  // [N,128]
    float* P1  = (float*)d_ws;                   // [N,128] scratch

    const int N = in_sizes[1];                   // norm has N elements
    const int E = in_sizes[2];

    // out accumulates scatter-sum -> must start at zero each call
    hipMemsetAsync(d_out, 0, (size_t)N * D * sizeof(float), stream);

    const int tilesN = (N + 15) / 16;
    ngcf_p1_gemm<<<(tilesN + 7) / 8, 256, 0, stream>>>(ego, W1, P1, N);

    const int tilesE = (E + 15) / 16;
    ngcf_edge_kernel<<<(tilesE + 7) / 8, 256, 0, stream>>>(ego, norm, src, dst,
                                                           W2, P1, out, E);

    ngcf_finalize<<<(N + 7) / 8, 256, 0, stream>>>(out, P1, N);
}